// INR1D_27736898797781
// MI455X (gfx1250) — compile-verified
//
#include <hip/hip_runtime.h>
#include <math.h>

typedef float v2f __attribute__((ext_vector_type(2)));
typedef float v8f __attribute__((ext_vector_type(8)));
typedef __attribute__((address_space(3))) float lds_f32;

#define OUTC 65536
#define CDIM 4096
#define NFEAT 256
#define HID 256

// Async global->LDS copy (gfx1250): per-lane 4B scatter, tracked by ASYNCcnt.
__device__ __forceinline__ void async_ld_b32(float* dst_lds, const float* src) {
    lds_f32* d = (lds_f32*)dst_lds;
    asm volatile("global_load_async_to_lds_b32 %0, %1, off" :: "v"(d), "v"(src) : "memory");
}
__device__ __forceinline__ void wait_async0() {
    asm volatile("s_wait_asynccnt 0" ::: "memory");
}

// ---------------- Kernel 1: ensemble distances + partial sums ----------------
#define QB 256
#define NR1 20
#define FP1 260

__global__ __launch_bounds__(256) void dist_kernel(const float* __restrict__ inp,
                                                   float* __restrict__ dists,
                                                   float* __restrict__ partial)
{
    __shared__ float F[NR1 * FP1];
    __shared__ float red[256];
    __shared__ int baseS;

    int tid = threadIdx.x;
    int qbase = blockIdx.x * QB;

    if (tid == 0) {
        float x0 = (2.f * (float)qbase + 1.f) / 65536.f - 1.f - 1.f / 65536.f;
        x0 = fmaxf(x0, -1.f + 1e-6f);
        float ix0 = (x0 + 1.f) * 0.5f * 4095.f;
        int b = (int)floorf(ix0);
        baseS = min(max(b, 0), CDIM - NR1);
    }
    __syncthreads();
    int base = baseS;

    // F[r][f] = fT[base+r][f] = inp[f*4096 + base + r]  (async, ASYNCcnt-tracked)
    for (int i = tid; i < NR1 * NFEAT; i += 256) {
        int f = i / NR1;
        int r = i - f * NR1;
        async_ld_b32(&F[r * FP1 + f], &inp[f * CDIM + base + r]);
    }
    wait_async0();
    __syncthreads();

    int q = qbase + tid;
    float coord = (2.f * (float)q + 1.f) / 65536.f - 1.f;

    float ixc = (coord + 1.f) * 0.5f * 4095.f;
    float fc  = floorf(ixc);
    float sc  = ixc - fc;
    int j0g = min(max((int)fc, 0), CDIM - 1);
    int j1g = min(j0g + 1, CDIM - 1);
    int j0 = j0g - base, j1 = j1g - base;

    int i0s[2], i1s[2];
    float tt[2];
    #pragma unroll
    for (int e = 0; e < 2; ++e) {
        float off = e ? 1.f : -1.f;
        float x = coord + off / 65536.f;
        x = fminf(fmaxf(x, -1.f + 1e-6f), 1.f - 1e-6f);
        float ix = (x + 1.f) * 0.5f * 4095.f;
        float fi = floorf(ix);
        tt[e] = ix - fi;
        int i0g = min(max((int)fi, 0), CDIM - 1);
        int i1g = min(i0g + 1, CDIM - 1);
        i0s[e] = i0g - base;
        i1s[e] = i1g - base;
    }

    float d0 = 0.f, d1 = 0.f;
    for (int f = 0; f < NFEAT; ++f) {
        const float* Fr = F + f;
        float c  = Fr[j0 * FP1] * (1.f - sc) + Fr[j1 * FP1] * sc;
        float a0 = Fr[i0s[0] * FP1] * (1.f - tt[0]) + Fr[i1s[0] * FP1] * tt[0];
        float a1 = Fr[i0s[1] * FP1] * (1.f - tt[1]) + Fr[i1s[1] * FP1] * tt[1];
        float u0 = a0 - c, u1 = a1 - c;
        d0 += u0 * u0;
        d1 += u1 * u1;
    }
    dists[q]        = d0;
    dists[OUTC + q] = d1;

    red[tid] = d0 + d1;
    __syncthreads();
    for (int s = 128; s > 0; s >>= 1) {
        if (tid < s) red[tid] += red[tid + s];
        __syncthreads();
    }
    if (tid == 0) partial[blockIdx.x] = red[0];
}

// ---------------- Kernel 2: deterministic serial sum ----------------
__global__ void sum_kernel(const float* __restrict__ partial, float* __restrict__ sum, int nb)
{
    if (threadIdx.x == 0 && blockIdx.x == 0) {
        float s = 0.f;
        for (int i = 0; i < nb; ++i) s += partial[i];
        sum[0] = s;
    }
}

// ---------------- Kernel 3: fused gather + 4-layer WMMA MLP + combine ----------------
#define QT 32
#define ROWS 64
#define PX 268      // layer-1 input pitch (266 + 2 zero pad)
#define PH 260      // hidden activation pitch
#define NR2 8

// One MLP layer on this wave's 16-row tile. Weights staged transposed in LDS
// (WT[c][k], pitch Kpad) via async-to-LDS, shared by all 4 waves per block.
__device__ __forceinline__ void wmma_layer(const float* Xl, int px, float* Yl, int py,
                                           const float* __restrict__ Wg,
                                           const float* __restrict__ Bg,
                                           float* WT, int Kreal, int Kpad, bool relu,
                                           int rbase, int colL, int half, int tid)
{
    for (int ct = 0; ct < 16; ++ct) {
        int n0 = ct * 16;
        __syncthreads();   // previous tile's readers are done
        for (int idx = tid; idx < (Kpad << 4); idx += 128) {
            int k = idx >> 4;
            int c = idx & 15;
            if (k < Kreal) async_ld_b32(&WT[c * Kpad + k], &Wg[k * HID + n0 + c]);
            else           WT[c * Kpad + k] = 0.f;           // exact-zero pad rows
        }
        if (ct < 15) __builtin_prefetch(Wg + n0 + 16, 0, 1); // next column tile
        wait_async0();
        __syncthreads();

        float bv = Bg[n0 + colL];
        v8f acc = { bv, bv, bv, bv, bv, bv, bv, bv };
        const float* xrow = Xl + (rbase + colL) * px + half * 2;
        const float* wrow = WT + colL * Kpad + half * 2;
        #pragma unroll 4
        for (int k0 = 0; k0 < Kpad; k0 += 4) {
            // A frag (16x4 f32): lanes 0-15 K={k0,k0+1}, lanes 16-31 K={k0+2,k0+3}
            v2f a = { xrow[k0], xrow[k0 + 1] };
            // B frag (4x16 f32): same K split, column = lane&15
            v2f b = { wrow[k0], wrow[k0 + 1] };
            acc = __builtin_amdgcn_wmma_f32_16x16x4_f32(false, a, false, b,
                                                        (short)0, acc, false, false);
        }
        #pragma unroll
        for (int v = 0; v < 8; ++v) {
            float y = acc[v];
            if (relu) y = fmaxf(y, 0.f);
            int row = rbase + v + half * 8;   // C/D layout: VGPR v -> M=v / M=v+8
            Yl[row * py + n0 + colL] = y;
        }
    }
}

__global__ __launch_bounds__(128) void mlp_kernel(
    const float* __restrict__ inp,
    const float* __restrict__ W1, const float* __restrict__ b1,
    const float* __restrict__ W2, const float* __restrict__ b2,
    const float* __restrict__ W3, const float* __restrict__ b3,
    const float* __restrict__ W4, const float* __restrict__ b4,
    const float* __restrict__ dists, const float* __restrict__ sumptr,
    float* __restrict__ out)
{
    __shared__ float bufA[ROWS * PX];
    __shared__ float bufB[ROWS * PH];
    __shared__ __align__(16) float scratch[16 * PX];  // WT (16*268) and F (8*260) time-share
    __shared__ int   s0a[2 * QT], s1a[2 * QT];
    __shared__ float tta[2 * QT], rela[2 * QT];
    __shared__ float w0a[QT], w1a[QT], corda[QT];
    __shared__ int baseS;

    float* WT = scratch;
    float* F  = scratch;

    int tid = threadIdx.x;
    int qbase = blockIdx.x * QT;

    if (tid == 0) {
        float x0 = (2.f * (float)qbase + 1.f) / 65536.f - 1.f - 1.f / 65536.f;
        x0 = fmaxf(x0, -1.f + 1e-6f);
        float ix0 = (x0 + 1.f) * 0.5f * 4095.f;
        int b = (int)floorf(ix0);
        baseS = min(max(b, 0), CDIM - NR2);
    }
    __syncthreads();
    int base = baseS;

    // async-stage the bilinear feature rows
    for (int i = tid; i < NR2 * NFEAT; i += 128) {
        int r = i & (NR2 - 1);
        int f = i >> 3;
        async_ld_b32(&F[r * FP1 + f], &inp[f * CDIM + base + r]);
    }

    if (tid < 2 * QT) {
        int e  = tid >> 5;
        int ql = tid & (QT - 1);
        int q  = qbase + ql;
        float coord = (2.f * (float)q + 1.f) / 65536.f - 1.f;
        float off = e ? 1.f : -1.f;
        float x = coord + off / 65536.f;
        x = fminf(fmaxf(x, -1.f + 1e-6f), 1.f - 1e-6f);
        float ix = (x + 1.f) * 0.5f * 4095.f;
        float fi = floorf(ix);
        float t  = ix - fi;
        int i0g = min(max((int)fi, 0), CDIM - 1);
        int i1g = min(i0g + 1, CDIM - 1);
        s0a[tid] = i0g - base;
        s1a[tid] = i1g - base;
        tta[tid] = t;
        float fc0 = (2.f * (float)i0g + 1.f) / 4096.f - 1.f;
        float fc1 = (2.f * (float)i1g + 1.f) / 4096.f - 1.f;
        float qc  = fc0 * (1.f - t) + fc1 * t;
        rela[tid] = (coord - qc) * 256.f;
        if (e == 0) {
            corda[ql] = coord;
            float mean = sumptr[0] * (1.f / 131072.f);
            float sig  = fmaxf(sqrtf(mean), 1e-6f);
            float den  = 2.f * sig * sig;
            float a0 = -dists[q] / den;
            float a1 = -dists[OUTC + q] / den;
            float m  = fmaxf(a0, a1);
            float e0 = expf(a0 - m), e1 = expf(a1 - m);
            float inv = 1.f / (e0 + e1);
            w0a[ql] = e0 * inv;
            w1a[ql] = e1 * inv;
        }
    }
    wait_async0();
    __syncthreads();

    // Assemble X0: rows = e*QT+ql ; cols [0,256) feat, [256,264) penc, 264 rel, 265 cell, 266-267 zero
    for (int idx = tid; idx < ROWS * PX; idx += 128) {
        int row = idx / PX;
        int col = idx - row * PX;
        int ql  = row & (QT - 1);
        float val;
        if (col < NFEAT) {
            float t = tta[row];
            val = F[s0a[row] * FP1 + col] * (1.f - t) + F[s1a[row] * FP1 + col] * t;
        } else if (col < 264) {
            int l = col - 256;
            float freq = 3.14159265358979323846f * (float)(1 << (l & 3));
            float s = corda[ql] * freq;
            val = (l < 4) ? sinf(s) : cosf(s);
        } else if (col == 264) {
            val = rela[row];
        } else if (col == 265) {
            val = 2.f / 65536.f;
        } else {
            val = 0.f;
        }
        bufA[idx] = val;
    }
    __syncthreads();

    int lane  = tid & 31;
    int wave  = tid >> 5;
    int colL  = lane & 15;
    int half  = lane >> 4;
    int rbase = wave * 16;

    wmma_layer(bufA, PX, bufB, PH, W1, b1, WT, 266, PX,  true,  rbase, colL, half, tid);
    wmma_layer(bufB, PH, bufA, PH, W2, b2, WT, HID, HID, true,  rbase, colL, half, tid);
    wmma_layer(bufA, PH, bufB, PH, W3, b3, WT, HID, HID, true,  rbase, colL, half, tid);
    wmma_layer(bufB, PH, bufA, PH, W4, b4, WT, HID, HID, false, rbase, colL, half, tid);
    __syncthreads();

    // Ensemble combine + reference reshape/permute scatter:
    // out_flat[(i%256)*65536 + j*256 + i/256] = w0*pred0[i][j] + w1*pred1[i][j]
    for (int idx = tid; idx < QT * NFEAT; idx += 128) {
        int ql = idx >> 8;
        int j  = idx & 255;
        float val = w0a[ql] * bufA[ql * PH + j] + w1a[ql] * bufA[(QT + ql) * PH + j];
        int i = qbase + ql;
        out[(i & 255) * 65536 + j * 256 + (i >> 8)] = val;
    }
}

extern "C" void kernel_launch(void* const* d_in, const int* in_sizes, int n_in,
                              void* d_out, int out_size, void* d_ws, size_t ws_size,
                              hipStream_t stream) {
    const float* inp = (const float*)d_in[0];
    const float* W1  = (const float*)d_in[1];
    const float* b1  = (const float*)d_in[2];
    const float* W2  = (const float*)d_in[3];
    const float* b2  = (const float*)d_in[4];
    const float* W3  = (const float*)d_in[5];
    const float* b3  = (const float*)d_in[6];
    const float* W4  = (const float*)d_in[7];
    const float* b4  = (const float*)d_in[8];
    float* out = (float*)d_out;

    float* ws      = (float*)d_ws;
    float* dists   = ws;                 // 2*65536 floats
    float* partial = ws + 2 * OUTC;      // 256 floats
    float* sum     = partial + 256;      // 1 float

    dist_kernel<<<OUTC / QB, 256, 0, stream>>>(inp, dists, partial);
    sum_kernel<<<1, 1, 0, stream>>>(partial, sum, OUTC / QB);
    mlp_kernel<<<OUTC / QT, 128, 0, stream>>>(inp, W1, b1, W2, b2, W3, b3, W4, b4,
                                              dists, sum, out);
}